// Integrate_feats_module_53300544143681
// MI455X (gfx1250) — compile-verified
//
#include <hip/hip_runtime.h>
#include <hip/hip_bf16.h>
#include <math.h>

typedef __attribute__((ext_vector_type(2))) float v2f;
typedef __attribute__((ext_vector_type(8))) float v8f;

// ---------------------------------------------------------------------------
// Kernel 1: weighted[b][hw] = sum_c W_cls[pids[b]][c] * feats[b][c][hw]
// bs=128, c=2048, hw=128.  Bandwidth-bound (134 MB streamed once).
// One block per b; 8 waves split the c-range; lane covers 4 hw via float4.
// ---------------------------------------------------------------------------
__global__ void __launch_bounds__(256)
weighted_pool_kernel(const float* __restrict__ feats,
                     const long long* __restrict__ pids,
                     const float* __restrict__ Wc,
                     float* __restrict__ weighted) {
    __shared__ float red[8 * 128];
    const int b    = blockIdx.x;
    const int tid  = threadIdx.x;
    const int wave = tid >> 5;
    const int lane = tid & 31;

    const long long pid = pids[b];
    const float* __restrict__ wrow  = Wc + (size_t)pid * 2048;
    const float* __restrict__ fbase = feats + (size_t)b * 2048 * 128 + lane * 4;

    float4 acc = make_float4(0.f, 0.f, 0.f, 0.f);
#pragma unroll 4
    for (int c = wave; c < 2048; c += 8) {
        // speculative HBM prefetch (emits global_prefetch_b8)
        __builtin_prefetch(fbase + (size_t)(c + 64) * 128, 0, 1);
        const float  cw = wrow[c];
        const float4 f  = *(const float4*)(fbase + (size_t)c * 128);
        acc.x = fmaf(cw, f.x, acc.x);
        acc.y = fmaf(cw, f.y, acc.y);
        acc.z = fmaf(cw, f.z, acc.z);
        acc.w = fmaf(cw, f.w, acc.w);
    }
    *(float4*)&red[wave * 128 + lane * 4] = acc;
    __syncthreads();

    if (tid < 128) {
        float s = 0.f;
#pragma unroll
        for (int wv = 0; wv < 8; ++wv) s += red[wv * 128 + tid];
        weighted[b * 128 + tid] = s;
    }
}

// ---------------------------------------------------------------------------
// Kernel 2: the GNN tail, algebraically collapsed:
//   u[b,:] = 0.25 * sum_{c=0..3} weighted[4b+c, :]          (32 x 128)
//   v      = elu(u @ W1) @ W2                               (two 32x128x128 GEMMs)
//   out[b,:] = v[b,:] / max(||v[b,:]||, 1e-12)
// Single 256-thread block, 8 waves. WMMA f32 16x16x4, K accumulated 4 at a time.
// Wave w owns N-strip [16w,16w+16); computes both M tiles (rows 0-15, 16-31).
// ---------------------------------------------------------------------------
#define LSTR 132   // LDS row stride (floats); 132%64=4 -> conflict-free strided rows

__global__ void __launch_bounds__(256)
gnn_wmma_kernel(const float* __restrict__ weighted,
                const float* __restrict__ W1,
                const float* __restrict__ W2,
                const long long* __restrict__ pids,
                float* __restrict__ out) {
    __shared__ float U[32 * LSTR];
    __shared__ float E[32 * LSTR];
    __shared__ float V[32 * LSTR];

    const int tid = threadIdx.x;

    // u = chunk mean (uniform trip count -> no divergence, EXEC stays full)
    for (int i = tid; i < 32 * 128; i += 256) {
        const int row = i >> 7, col = i & 127;
        const float s = weighted[(4 * row + 0) * 128 + col]
                      + weighted[(4 * row + 1) * 128 + col]
                      + weighted[(4 * row + 2) * 128 + col]
                      + weighted[(4 * row + 3) * 128 + col];
        U[row * LSTR + col] = 0.25f * s;
    }
    __syncthreads();

    const int wave = tid >> 5;
    const int lane = tid & 31;
    const int half = lane >> 4;   // selects K+2 (A/B) and M+8 (C/D)
    const int lm   = lane & 15;
    const int n0   = wave * 16;
    const int n    = n0 + lm;

    // ---------------- GEMM1: E = elu(U @ W1) ----------------
    {
        v8f acc0 = {}; v8f acc1 = {};
        for (int k = 0; k < 128; k += 4) {
            const int kb = k + 2 * half;
            v2f a0, a1, bb;
            a0.x = U[lm * LSTR + kb];          a0.y = U[lm * LSTR + kb + 1];
            a1.x = U[(16 + lm) * LSTR + kb];   a1.y = U[(16 + lm) * LSTR + kb + 1];
            bb.x = W1[kb * 128 + n];           bb.y = W1[(kb + 1) * 128 + n];
            acc0 = __builtin_amdgcn_wmma_f32_16x16x4_f32(false, a0, false, bb,
                                                         (short)0, acc0, false, false);
            acc1 = __builtin_amdgcn_wmma_f32_16x16x4_f32(false, a1, false, bb,
                                                         (short)0, acc1, false, false);
        }
#pragma unroll
        for (int r = 0; r < 8; ++r) {
            const int row = r + 8 * half;
            const float x0 = acc0[r];
            const float x1 = acc1[r];
            E[row * LSTR + n]        = x0 > 0.f ? x0 : (__expf(x0) - 1.f);
            E[(16 + row) * LSTR + n] = x1 > 0.f ? x1 : (__expf(x1) - 1.f);
        }
    }
    __syncthreads();

    // ---------------- GEMM2: V = E @ W2 ----------------
    {
        v8f acc0 = {}; v8f acc1 = {};
        for (int k = 0; k < 128; k += 4) {
            const int kb = k + 2 * half;
            v2f a0, a1, bb;
            a0.x = E[lm * LSTR + kb];          a0.y = E[lm * LSTR + kb + 1];
            a1.x = E[(16 + lm) * LSTR + kb];   a1.y = E[(16 + lm) * LSTR + kb + 1];
            bb.x = W2[kb * 128 + n];           bb.y = W2[(kb + 1) * 128 + n];
            acc0 = __builtin_amdgcn_wmma_f32_16x16x4_f32(false, a0, false, bb,
                                                         (short)0, acc0, false, false);
            acc1 = __builtin_amdgcn_wmma_f32_16x16x4_f32(false, a1, false, bb,
                                                         (short)0, acc1, false, false);
        }
#pragma unroll
        for (int r = 0; r < 8; ++r) {
            const int row = r + 8 * half;
            V[row * LSTR + n]        = acc0[r];
            V[(16 + row) * LSTR + n] = acc1[r];
        }
    }
    __syncthreads();

    // ---------------- row-wise L2 normalize + outputs ----------------
    if (tid < 32) {
        float ss = 0.f;
        for (int j = 0; j < 128; ++j) { const float v = V[tid * LSTR + j]; ss = fmaf(v, v, ss); }
        const float nrm  = fmaxf(sqrtf(ss), 1e-12f);
        const float rinv = 1.f / nrm;
        for (int j = 0; j < 128; ++j) out[tid * 128 + j] = V[tid * LSTR + j] * rinv;
        // integrate_pids = pids[::4], appended after the 32*128 floats
        long long* po = (long long*)(out + 32 * 128);
        po[tid] = pids[4 * tid];
    }
}

// ---------------------------------------------------------------------------
// Launch: inputs are (feats, pids, W_cls, W1, W2, num_same_id)
// ---------------------------------------------------------------------------
extern "C" void kernel_launch(void* const* d_in, const int* in_sizes, int n_in,
                              void* d_out, int out_size, void* d_ws, size_t ws_size,
                              hipStream_t stream) {
    const float*     feats = (const float*)d_in[0];
    const long long* pids  = (const long long*)d_in[1];
    const float*     Wc    = (const float*)d_in[2];
    const float*     W1    = (const float*)d_in[3];
    const float*     W2    = (const float*)d_in[4];
    float*           out   = (float*)d_out;
    float*           wgt   = (float*)d_ws;   // 128*128 floats = 64 KB scratch

    weighted_pool_kernel<<<128, 256, 0, stream>>>(feats, pids, Wc, wgt);
    gnn_wmma_kernel<<<1, 256, 0, stream>>>(wgt, W1, W2, pids, out);
}